// GTN_g2_40132174414169
// MI455X (gfx1250) — compile-verified
//
#include <hip/hip_runtime.h>
#include <math.h>

typedef __attribute__((ext_vector_type(2))) float v2f;
typedef __attribute__((ext_vector_type(8))) float v8f;

#define Nn   4096
#define INF_ 3.402823466e+38f
#define Dd   256
#define Hh   8
#define HDd  32
#define Ll   2
#define Ee   131072
#define Pp   200000
#define INc  128

__device__ __forceinline__ v8f wmma4(v2f a, v2f b, v8f c) {
  // V_WMMA_F32_16X16X4_F32 : D = A(16x4) * B(4x16) + C(16x16), fp32 exact
  return __builtin_amdgcn_wmma_f32_16x16x4_f32(
      false, a, false, b, (short)0, c, false, false);
}

__device__ __forceinline__ float rsum16(float v) {
  for (int m = 8; m >= 1; m >>= 1) v += __shfl_xor(v, m, 16);
  return v;
}
__device__ __forceinline__ float rmax16(float v) {
  for (int m = 8; m >= 1; m >>= 1) v = fmaxf(v, __shfl_xor(v, m, 16));
  return v;
}

// ------- GEMM: C[M][Nout] = A[M][K] * W[Nout][K]^T + bias --------------------
// One wave computes a 16x64 tile: 4 accumulators share each A operand (1 A load
// + 4 B loads -> 4 WMMAs per K-step). K is a compile-time constant so the fully
// unrolled loop addresses operands with immediate offsets: no VALU address
// chains, loads batch into clauses with staggered s_wait_loadcnt.
template <int KK>
__global__ __launch_bounds__(32)
void gemm_atb(const float* __restrict__ A, const float* __restrict__ W,
              const float* __restrict__ bias, float* __restrict__ C,
              int M, int Nout) {
  const int tid  = threadIdx.x;
  const int m    = tid & 15;
  const int half = tid >> 4;
  const int tbn  = blockIdx.x * 64;
  const int tbm  = blockIdx.y * 16;

  const float* arow = A + (size_t)(tbm + m) * KK + half * 2;
  const float* w0   = W + (size_t)(tbn + m) * KK + half * 2;
  const float* w1   = w0 + (size_t)16 * KK;
  const float* w2   = w1 + (size_t)16 * KK;
  const float* w3   = w2 + (size_t)16 * KK;

  v8f c0 = {}, c1 = {}, c2 = {}, c3 = {};
#pragma unroll
  for (int k = 0; k < KK; k += 4) {
    v2f a  = { arow[k], arow[k + 1] };
    v2f b0 = { w0[k],   w0[k + 1] };
    v2f b1 = { w1[k],   w1[k + 1] };
    v2f b2 = { w2[k],   w2[k + 1] };
    v2f b3 = { w3[k],   w3[k + 1] };
    c0 = wmma4(a, b0, c0);
    c1 = wmma4(a, b1, c1);
    c2 = wmma4(a, b2, c2);
    c3 = wmma4(a, b3, c3);
  }
  for (int r = 0; r < 8; ++r) {
    const int row = tbm + r + 8 * half;
    float* crow = C + (size_t)row * Nout + tbn + m;
    if (bias) {
      const float* brow = bias + tbn + m;
      crow[0]  = c0[r] + brow[0];
      crow[16] = c1[r] + brow[16];
      crow[32] = c2[r] + brow[32];
      crow[48] = c3[r] + brow[48];
    } else {
      crow[0] = c0[r]; crow[16] = c1[r]; crow[32] = c2[r]; crow[48] = c3[r];
    }
  }
}

// ---------------- flash attention: one wave per (16 query rows, head) --------
__global__ __launch_bounds__(32)
void attn_kernel(const float* __restrict__ qkv, float* __restrict__ obuf) {
  const int tid  = threadIdx.x;
  const int m    = tid & 15;
  const int half = tid >> 4;
  const int qb   = blockIdx.x;   // N/16 blocks
  const int h    = blockIdx.y;   // H heads
  const float scale = 0.1767766952966369f; // 1/sqrt(32)

  const float* qbase = qkv + (size_t)(qb * 16) * (3 * Dd) + h * HDd;

  __shared__ float pLds[16 * 16];

  // Q tile is loop-invariant: keep its 8 operand pairs in registers.
  v2f qa[8];
#pragma unroll
  for (int kk = 0; kk < HDd; kk += 4) {
    const int k0 = kk + half * 2;
    qa[kk >> 2] = v2f{ qbase[m * (3 * Dd) + k0], qbase[m * (3 * Dd) + k0 + 1] };
  }

  v8f o0 = {}, o1 = {};
  float mrow[8], lrow[8];
  for (int r = 0; r < 8; ++r) { mrow[r] = -INF_; lrow[r] = 0.f; }

  for (int kb = 0; kb < Nn / 16; ++kb) {
    const float* kbase = qkv + (size_t)(kb * 16) * (3 * Dd) + Dd + h * HDd;
    const float* vbase = qkv + (size_t)(kb * 16) * (3 * Dd) + 2 * Dd + h * HDd;

    // S = Q * K^T
    v8f s = {};
#pragma unroll
    for (int kk = 0; kk < HDd; kk += 4) {
      const int k0 = kk + half * 2;
      v2f b = { kbase[m * (3 * Dd) + k0], kbase[m * (3 * Dd) + k0 + 1] };
      s = wmma4(qa[kk >> 2], b, s);
    }

    // online softmax, per-row stats replicated over each 16-lane half
    float alpha[8];
#pragma unroll
    for (int r = 0; r < 8; ++r) {
      float sv = s[r] * scale;
      float rm = rmax16(sv);
      float mn = fmaxf(mrow[r], rm);
      float p  = __expf(sv - mn);
      float co = __expf(mrow[r] - mn);
      float rs = rsum16(p);
      lrow[r] = lrow[r] * co + rs;
      mrow[r] = mn;
      alpha[r] = co;
      s[r] = p;
    }
#pragma unroll
    for (int r = 0; r < 8; ++r) { o0[r] *= alpha[r]; o1[r] *= alpha[r]; }

    // stage P through LDS: convert C-layout -> A-layout
    __syncthreads();
#pragma unroll
    for (int r = 0; r < 8; ++r) pLds[(r + 8 * half) * 16 + m] = s[r];
    __syncthreads();

    // O += P * V   (two 16x16 tiles cover HD=32, shared A operand)
#pragma unroll
    for (int kk = 0; kk < 16; kk += 4) {
      const int k0 = kk + half * 2;
      v2f a  = { pLds[m * 16 + k0], pLds[m * 16 + k0 + 1] };
      v2f b0 = { vbase[k0 * (3 * Dd) + m],      vbase[(k0 + 1) * (3 * Dd) + m] };
      v2f b1 = { vbase[k0 * (3 * Dd) + 16 + m], vbase[(k0 + 1) * (3 * Dd) + 16 + m] };
      o0 = wmma4(a, b0, o0);
      o1 = wmma4(a, b1, o1);
    }
  }

  float* ob = obuf + (size_t)(qb * 16) * Dd + h * HDd;
#pragma unroll
  for (int r = 0; r < 8; ++r) {
    const int row = r + 8 * half;
    const float inv = 1.0f / lrow[r];
    ob[row * Dd + m]      = o0[r] * inv;
    ob[row * Dd + 16 + m] = o1[r] * inv;
  }
}

// ---------------- segment sum of edge weights ---------------------------------
__global__ void zerof(float* p, int n) {
  int i = blockIdx.x * blockDim.x + threadIdx.x;
  if (i < n) p[i] = 0.f;
}
__global__ void seg_sum(const float* __restrict__ ew, const int* __restrict__ src,
                        float* __restrict__ out, int E) {
  int i = blockIdx.x * blockDim.x + threadIdx.x;
  if (i < E) atomicAdd(&out[src[i]], ew[i]);
}

// ---------------- h = LN(h + add * ws[row]) * g + b ---------------------------
__global__ __launch_bounds__(256)
void residual_ln(float* __restrict__ h, const float* __restrict__ add,
                 const float* __restrict__ wsn,
                 const float* __restrict__ g, const float* __restrict__ b) {
  const int row = blockIdx.x;
  const int c = threadIdx.x;
  const float wsv = wsn ? wsn[row] : 1.0f;
  float t = h[(size_t)row * Dd + c] + add[(size_t)row * Dd + c] * wsv;

  __shared__ float red[Dd];
  red[c] = t; __syncthreads();
  for (int s = Dd / 2; s > 0; s >>= 1) { if (c < s) red[c] += red[c + s]; __syncthreads(); }
  const float mean = red[0] / (float)Dd;
  __syncthreads();
  const float d = t - mean;
  red[c] = d * d; __syncthreads();
  for (int s = Dd / 2; s > 0; s >>= 1) { if (c < s) red[c] += red[c + s]; __syncthreads(); }
  const float var = red[0] / (float)Dd;
  const float rs = rsqrtf(var + 1e-5f);
  h[(size_t)row * Dd + c] = d * rs * g[c] + b[c];
}

// ---------------- h_combined = h + h_proj -------------------------------------
__global__ void add_kernel(const float* a, const float* b, float* o, int n) {
  int i = blockIdx.x * blockDim.x + threadIdx.x;
  if (i < n) o[i] = a[i] + b[i];
}

// ---------------- link predictor: 16 pairs per wave ---------------------------
__global__ __launch_bounds__(32)
void predict16(const float* __restrict__ hc,
               const int* __restrict__ si, const int* __restrict__ di,
               const float* __restrict__ pw1, const float* __restrict__ pb1,
               const float* __restrict__ pw2, const float* __restrict__ pb2,
               float* __restrict__ out) {
  const int tid  = threadIdx.x;
  const int base = blockIdx.x * 16;

  __shared__ float z[16 * Dd]; // 16KB
  for (int idx = tid; idx < 16 * Dd; idx += 32) {
    const int row = idx >> 8, col = idx & (Dd - 1);
    const int rs = si[base + row], rd = di[base + row];
    z[idx] = hc[(size_t)rs * Dd + col] * hc[(size_t)rd * Dd + col];
  }
  __syncthreads();

  const int m = tid & 15;
  const int half = tid >> 4;
  float acc[8];
#pragma unroll
  for (int r = 0; r < 8; ++r) acc[r] = 0.f;

  for (int jt = 0; jt < 8; ++jt) {          // 128 hidden units / 16
    v8f c = {};
    const float* wrow = pw1 + (size_t)(jt * 16 + m) * Dd + half * 2;
    const float* zrow = z + m * Dd + half * 2;
#pragma unroll
    for (int k = 0; k < Dd; k += 4) {
      v2f a = { zrow[k], zrow[k + 1] };
      v2f b = { wrow[k], wrow[k + 1] };
      c = wmma4(a, b, c);
    }
    const float bias1 = pb1[jt * 16 + m];
    const float w2 = pw2[jt * 16 + m];
#pragma unroll
    for (int r = 0; r < 8; ++r) {
      float v = c[r] + bias1;
      v = v > 0.f ? v : 0.2f * v;           // leaky relu, slope 0.2
      acc[r] += v * w2;
    }
  }
#pragma unroll
  for (int r = 0; r < 8; ++r) {
    const float sr = rsum16(acc[r]);
    if (m == 0) out[base + r + 8 * half] = sr + pb2[0];
  }
}

extern "C" void kernel_launch(void* const* d_in, const int* in_sizes, int n_in,
                              void* d_out, int out_size, void* d_ws, size_t ws_size,
                              hipStream_t stream) {
  const float* x       = (const float*)d_in[0];
  const float* edge_w  = (const float*)d_in[1];
  const int*   src     = (const int*)d_in[2];
  // d_in[3] = dst (unused: scatter-assign + row-sum == segment-sum over src)
  const int*   pos_src = (const int*)d_in[4];
  const int*   pos_dst = (const int*)d_in[5];
  const int*   neg_src = (const int*)d_in[6];
  const int*   neg_dst = (const int*)d_in[7];
  const float* proj_w  = (const float*)d_in[8];
  const float* proj_b  = (const float*)d_in[9];
  const float* in_w    = (const float*)d_in[10];
  const float* in_b    = (const float*)d_in[11];
  const float* out_w   = (const float*)d_in[12];
  const float* out_b   = (const float*)d_in[13];
  const float* ff_w    = (const float*)d_in[14];
  const float* ff_b    = (const float*)d_in[15];
  const float* g1      = (const float*)d_in[16];
  const float* b1      = (const float*)d_in[17];
  const float* g2      = (const float*)d_in[18];
  const float* b2      = (const float*)d_in[19];
  const float* pw1     = (const float*)d_in[20];
  const float* pb1     = (const float*)d_in[21];
  const float* pw2     = (const float*)d_in[22];
  const float* pb2     = (const float*)d_in[23];

  float* W      = (float*)d_ws;
  float* h      = W;                      // N*D
  float* hproj  = W + 1048576;            // N*D
  float* qkv    = W + 2097152;            // N*3D
  float* obuf   = W + 5242880;            // N*D
  float* scr    = W + 6291456;            // N*D
  float* wsn    = W + 7340032;            // N

  float* outp = (float*)d_out;
  float* pos  = outp;
  float* neg  = outp + Pp;
  float* hc   = outp + 2 * Pp;            // h_combined lives in d_out

  // h_proj = x @ proj_w^T + proj_b ; h = h_proj
  gemm_atb<INc><<<dim3(Dd / 64, Nn / 16), 32, 0, stream>>>(x, proj_w, proj_b, hproj, Nn, Dd);
  hipMemcpyAsync(h, hproj, (size_t)Nn * Dd * sizeof(float), hipMemcpyDeviceToDevice, stream);

  for (int l = 0; l < Ll; ++l) {
    // qkv = h @ in_w[l]^T + in_b[l]
    gemm_atb<Dd><<<dim3(3 * Dd / 64, Nn / 16), 32, 0, stream>>>(
        h, in_w + (size_t)l * 3 * Dd * Dd, in_b + (size_t)l * 3 * Dd, qkv, Nn, 3 * Dd);
    // flash attention -> obuf
    attn_kernel<<<dim3(Nn / 16, Hh), 32, 0, stream>>>(qkv, obuf);
    // out projection -> scr
    gemm_atb<Dd><<<dim3(Dd / 64, Nn / 16), 32, 0, stream>>>(
        obuf, out_w + (size_t)l * Dd * Dd, out_b + (size_t)l * Dd, scr, Nn, Dd);
    // ws = segment_sum(edge_w[l], src[l])
    zerof<<<(Nn + 255) / 256, 256, 0, stream>>>(wsn, Nn);
    seg_sum<<<(Ee + 255) / 256, 256, 0, stream>>>(edge_w + (size_t)l * Ee,
                                                  src + (size_t)l * Ee, wsn, Ee);
    // h = LN(h + attn*ws)
    residual_ln<<<Nn, Dd, 0, stream>>>(h, scr, wsn, g1 + (size_t)l * Dd, b1 + (size_t)l * Dd);
    // ff -> scr ; h = LN(h + ff)
    gemm_atb<Dd><<<dim3(Dd / 64, Nn / 16), 32, 0, stream>>>(
        h, ff_w + (size_t)l * Dd * Dd, ff_b + (size_t)l * Dd, scr, Nn, Dd);
    residual_ln<<<Nn, Dd, 0, stream>>>(h, scr, nullptr, g2 + (size_t)l * Dd, b2 + (size_t)l * Dd);
  }

  // h_combined = h + h_proj (directly into d_out tail)
  add_kernel<<<(Nn * Dd + 255) / 256, 256, 0, stream>>>(h, hproj, hc, Nn * Dd);

  // scores
  predict16<<<Pp / 16, 32, 0, stream>>>(hc, pos_src, pos_dst, pw1, pb1, pw2, pb2, pos);
  predict16<<<Pp / 16, 32, 0, stream>>>(hc, neg_src, neg_dst, pw1, pb1, pw2, pb2, neg);
}